// MultiModalContrastiveLoss_57397942944133
// MI455X (gfx1250) — compile-verified
//
#include <hip/hip_runtime.h>
#include <math.h>

#define BB    16
#define CC    256
#define HWN   2304          // 48*48
#define TILES 9             // HWN / 256
#define NBLK  (BB*TILES)    // 144

typedef float v2f __attribute__((ext_vector_type(2)));
typedef float v8f __attribute__((ext_vector_type(8)));

// workspace layout (float offsets)
enum : int {
    OFF_INV_IMG = 0,
    OFF_INV_PT  = OFF_INV_IMG + BB*HWN,   //  36864
    OFF_INV_BEV = OFF_INV_PT  + BB*HWN,   //  73728
    OFF_DIAG_IB = OFF_INV_BEV + BB*HWN,   // 110592
    OFF_DIAG_PB = OFF_DIAG_IB + BB*HWN,   // 147456
    OFF_BEVSUM  = OFF_DIAG_PB + BB*HWN,   // 184320
    OFF_PDI     = OFF_BEVSUM  + BB*CC,    // 188416
    OFF_PDP     = OFF_PDI + NBLK,
    OFF_PNEG    = OFF_PDP + NBLK,
    WS_FLOATS   = OFF_PNEG + NBLK
};

__device__ __forceinline__ float sigmoidf_(float x) { return 1.0f / (1.0f + expf(-x)); }

// ---------------------------------------------------------------------------
// K1: per-position channel reductions: norms of img/pt/bev, dot(img,bev),
//     dot(pt,bev). 128 threads x 2 adjacent n-positions (b64 loads,
//     coalesced over n, the innermost dim).
// ---------------------------------------------------------------------------
__global__ __launch_bounds__(128)
void k1_norms_diags(const float* __restrict__ img,
                    const float* __restrict__ pt,
                    const float* __restrict__ bev,
                    float* __restrict__ ws)
{
    const int blk = blockIdx.x;                         // 0..143
    const int b   = blk / TILES;
    const int n0  = (blk % TILES) * 256 + threadIdx.x * 2;
    const size_t base2 = ((size_t)b * CC * HWN + n0) >> 1;   // float2 index

    const v2f* __restrict__ img2 = (const v2f*)img;
    const v2f* __restrict__ pt2  = (const v2f*)pt;
    const v2f* __restrict__ bev2 = (const v2f*)bev;

    float ssi0 = 0.f, ssp0 = 0.f, ssb0 = 0.f, dib0 = 0.f, dpb0 = 0.f;
    float ssi1 = 0.f, ssp1 = 0.f, ssb1 = 0.f, dib1 = 0.f, dpb1 = 0.f;
#pragma unroll 4
    for (int c = 0; c < CC; ++c) {
        const size_t idx = base2 + (size_t)c * (HWN / 2);
        const v2f vi = img2[idx];
        const v2f vp = pt2[idx];
        const v2f vb = bev2[idx];
        ssi0 = fmaf(vi.x, vi.x, ssi0);  ssi1 = fmaf(vi.y, vi.y, ssi1);
        ssp0 = fmaf(vp.x, vp.x, ssp0);  ssp1 = fmaf(vp.y, vp.y, ssp1);
        ssb0 = fmaf(vb.x, vb.x, ssb0);  ssb1 = fmaf(vb.y, vb.y, ssb1);
        dib0 = fmaf(vi.x, vb.x, dib0);  dib1 = fmaf(vi.y, vb.y, dib1);
        dpb0 = fmaf(vp.x, vb.x, dpb0);  dpb1 = fmaf(vp.y, vb.y, dpb1);
    }
    const float inv_i0 = 1.0f / fmaxf(sqrtf(ssi0), 1e-12f);
    const float inv_i1 = 1.0f / fmaxf(sqrtf(ssi1), 1e-12f);
    const float inv_p0 = 1.0f / fmaxf(sqrtf(ssp0), 1e-12f);
    const float inv_p1 = 1.0f / fmaxf(sqrtf(ssp1), 1e-12f);
    const float inv_b0 = 1.0f / fmaxf(sqrtf(ssb0), 1e-12f);
    const float inv_b1 = 1.0f / fmaxf(sqrtf(ssb1), 1e-12f);
    const int bn = b * HWN + n0;
    ((v2f*)(ws + OFF_INV_IMG))[bn >> 1] = (v2f){inv_i0, inv_i1};
    ((v2f*)(ws + OFF_INV_PT ))[bn >> 1] = (v2f){inv_p0, inv_p1};
    ((v2f*)(ws + OFF_INV_BEV))[bn >> 1] = (v2f){inv_b0, inv_b1};
    const float d_ib0 = dib0 * inv_i0 * inv_b0;
    const float d_ib1 = dib1 * inv_i1 * inv_b1;
    const float d_pb0 = dpb0 * inv_p0 * inv_b0;
    const float d_pb1 = dpb1 * inv_p1 * inv_b1;
    ((v2f*)(ws + OFF_DIAG_IB))[bn >> 1] = (v2f){d_ib0, d_ib1};
    ((v2f*)(ws + OFF_DIAG_PB))[bn >> 1] = (v2f){d_pb0, d_pb1};

    // fixed-order block partials of the diagonal sums (for pos term)
    __shared__ float r1[128], r2[128];
    r1[threadIdx.x] = d_ib0 + d_ib1;
    r2[threadIdx.x] = d_pb0 + d_pb1;
    __syncthreads();
    for (int s = 64; s > 0; s >>= 1) {
        if (threadIdx.x < s) {
            r1[threadIdx.x] += r1[threadIdx.x + s];
            r2[threadIdx.x] += r2[threadIdx.x + s];
        }
        __syncthreads();
    }
    if (threadIdx.x == 0) {
        ws[OFF_PDI + blk] = r1[0];
        ws[OFF_PDP + blk] = r2[0];
    }
}

// ---------------------------------------------------------------------------
// K2: bevsum[b][c] = sum_n bev[b,c,n] * inv_bev[b,n].  One block per (b,c);
//     bev is L2-resident after K1 (113MB < 192MB L2). float2 streams.
// ---------------------------------------------------------------------------
__global__ __launch_bounds__(256)
void k2_bevsum(const float* __restrict__ bev, float* __restrict__ ws)
{
    const int b = blockIdx.x / CC;
    const int c = blockIdx.x % CC;
    const v2f* __restrict__ row2 = (const v2f*)(bev + (size_t)(b * CC + c) * HWN);
    const v2f* __restrict__ inv2 = (const v2f*)(ws + OFF_INV_BEV + b * HWN);
    float s = 0.f;
    for (int n = threadIdx.x; n < HWN / 2; n += 256) {
        const v2f rv = row2[n];
        const v2f iv = inv2[n];
        s = fmaf(rv.x, iv.x, s);
        s = fmaf(rv.y, iv.y, s);
    }
    __shared__ float r[256];
    r[threadIdx.x] = s;
    __syncthreads();
    for (int t = 128; t > 0; t >>= 1) {
        if (threadIdx.x < t) r[threadIdx.x] += r[threadIdx.x + t];
        __syncthreads();
    }
    if (threadIdx.x == 0) ws[OFF_BEVSUM + b * CC + c] = r[0];
}

// ---------------------------------------------------------------------------
// K3: row-sum factorization: sum_m <x_n, bev_m> = <x_n, bevsum_b>.
//     neg = (rowsum - diag)/(N-1); accumulate sigmoid(neg) block partials.
//     128 threads x 2 adjacent n-positions (b64 loads).
// ---------------------------------------------------------------------------
__global__ __launch_bounds__(128)
void k3_neg(const float* __restrict__ img, const float* __restrict__ pt,
            float* __restrict__ ws)
{
    const int blk = blockIdx.x;
    const int b   = blk / TILES;
    const int n0  = (blk % TILES) * 256 + threadIdx.x * 2;

    __shared__ float bs[CC];
    bs[threadIdx.x]       = ws[OFF_BEVSUM + b * CC + threadIdx.x];
    bs[threadIdx.x + 128] = ws[OFF_BEVSUM + b * CC + threadIdx.x + 128];
    __syncthreads();

    const size_t base2 = ((size_t)b * CC * HWN + n0) >> 1;
    const v2f* __restrict__ img2 = (const v2f*)img;
    const v2f* __restrict__ pt2  = (const v2f*)pt;

    float si0 = 0.f, sp0 = 0.f, si1 = 0.f, sp1 = 0.f;
#pragma unroll 4
    for (int c = 0; c < CC; ++c) {
        const size_t idx = base2 + (size_t)c * (HWN / 2);
        const v2f vi = img2[idx];
        const v2f vp = pt2[idx];
        const float w = bs[c];
        si0 = fmaf(vi.x, w, si0);  si1 = fmaf(vi.y, w, si1);
        sp0 = fmaf(vp.x, w, sp0);  sp1 = fmaf(vp.y, w, sp1);
    }
    const int bn = b * HWN + n0;
    const float inv_nm1 = 1.0f / (float)(HWN - 1);
    const v2f invi = ((const v2f*)(ws + OFF_INV_IMG))[bn >> 1];
    const v2f invp = ((const v2f*)(ws + OFF_INV_PT ))[bn >> 1];
    const v2f dib  = ((const v2f*)(ws + OFF_DIAG_IB))[bn >> 1];
    const v2f dpb  = ((const v2f*)(ws + OFF_DIAG_PB))[bn >> 1];
    const float neg_i0 = (si0 * invi.x - dib.x) * inv_nm1;
    const float neg_i1 = (si1 * invi.y - dib.y) * inv_nm1;
    const float neg_p0 = (sp0 * invp.x - dpb.x) * inv_nm1;
    const float neg_p1 = (sp1 * invp.y - dpb.y) * inv_nm1;
    const float sig = sigmoidf_(neg_i0) + sigmoidf_(neg_i1)
                    + sigmoidf_(neg_p0) + sigmoidf_(neg_p1);

    __shared__ float r[128];
    r[threadIdx.x] = sig;
    __syncthreads();
    for (int t = 64; t > 0; t >>= 1) {
        if (threadIdx.x < t) r[threadIdx.x] += r[threadIdx.x + t];
        __syncthreads();
    }
    if (threadIdx.x == 0) ws[OFF_PNEG + blk] = r[0];
}

// ---------------------------------------------------------------------------
// K4: global + cross losses via fp32 WMMA (16x16x4_F32, K=256) Gram matrices,
//     then combine everything into (g, l, c, total). Single workgroup.
// ---------------------------------------------------------------------------
__global__ __launch_bounds__(256)
void k4_final(const float* __restrict__ ig_raw, const float* __restrict__ pg_raw,
              const float* __restrict__ tg_raw, const float* __restrict__ lsc,
              const float* __restrict__ ws, float* __restrict__ out)
{
    __shared__ float nig[16][CC], npg[16][CC], ntg[16][CC];  // 48 KB
    __shared__ float red[16][16];
    __shared__ float Git[16][16], Gip[16][16], Gpt[16][16];
    __shared__ float res[5][16];

    const int tid    = threadIdx.x;
    const int r      = tid >> 4;       // row 0..15
    const int lane16 = tid & 15;

    // row-normalize the three 16x256 global tensors into LDS
    const float* srcs[3] = { ig_raw, pg_raw, tg_raw };
    float (*dsts[3])[CC] = { nig, npg, ntg };
    for (int t = 0; t < 3; ++t) {
        const float* __restrict__ x = srcs[t];
        float ss = 0.f;
        for (int j = lane16; j < CC; j += 16) {
            const float v = x[r * CC + j];
            ss = fmaf(v, v, ss);
        }
        red[r][lane16] = ss;
        __syncthreads();
        if (lane16 == 0) {
            float tot = 0.f;
            for (int j = 0; j < 16; ++j) tot += red[r][j];
            red[r][0] = 1.0f / fmaxf(sqrtf(tot), 1e-12f);
        }
        __syncthreads();
        const float inv = red[r][0];
        for (int j = lane16; j < CC; j += 16)
            dsts[t][r][j] = x[r * CC + j] * inv;
        __syncthreads();
    }

    const float scale = expf(lsc[0]);

    // --- WMMA Gram matrices: 16x16 outputs, K=256, fp32 end-to-end ---
    // A layout (16x4): lane row = lane&15; VGPR0/1 hold K = koff, koff+1
    // with koff = 2*(lane>=16). B (4x16) loads symmetrically for X @ Y^T.
    // All 8 waves execute with full EXEC (redundantly); wave 0 stores D.
    const int l    = tid & 31;
    const int row  = l & 15;
    const int koff = (l >> 4) ? 2 : 0;
    v8f acc_it = {}; v8f acc_ip = {}; v8f acc_pt = {};
    for (int k0 = 0; k0 < CC; k0 += 4) {
        const int k = k0 + koff;
        v2f vi = { nig[row][k], nig[row][k + 1] };
        v2f vp = { npg[row][k], npg[row][k + 1] };
        v2f vt = { ntg[row][k], ntg[row][k + 1] };
        acc_it = __builtin_amdgcn_wmma_f32_16x16x4_f32(false, vi, false, vt, (short)0, acc_it, false, false);
        acc_ip = __builtin_amdgcn_wmma_f32_16x16x4_f32(false, vi, false, vp, (short)0, acc_ip, false, false);
        acc_pt = __builtin_amdgcn_wmma_f32_16x16x4_f32(false, vp, false, vt, (short)0, acc_pt, false, false);
    }
    if (tid < 32) {
        // C/D layout: VGPR rr, lane l -> M = rr + 8*(l>=16), N = l&15
        const int mbase = (l >> 4) ? 8 : 0;
#pragma unroll
        for (int rr = 0; rr < 8; ++rr) {
            Git[mbase + rr][row] = acc_it[rr] * scale;
            Gip[mbase + rr][row] = acc_ip[rr] * scale;
            Gpt[mbase + rr][row] = acc_pt[rr] * scale;
        }
    }
    __syncthreads();

    // stable logsumexp cross-entropies with diagonal labels
    if (tid < 16) {
        const int i = tid;
        auto lse_row = [&](float (*G)[16]) -> float {
            float m = -1e30f;
            for (int j = 0; j < 16; ++j) m = fmaxf(m, G[i][j]);
            float s = 0.f;
            for (int j = 0; j < 16; ++j) s += expf(G[i][j] - m);
            return m + logf(s);
        };
        auto lse_col = [&](float (*G)[16]) -> float {
            float m = -1e30f;
            for (int j = 0; j < 16; ++j) m = fmaxf(m, G[j][i]);
            float s = 0.f;
            for (int j = 0; j < 16; ++j) s += expf(G[j][i] - m);
            return m + logf(s);
        };
        res[0][i] = lse_row(Git) - Git[i][i];   // loss_it terms
        res[1][i] = lse_col(Git) - Git[i][i];   // loss_ti terms
        res[2][i] = lse_row(Gip) - Gip[i][i];   // loss_ip terms
        res[3][i] = lse_col(Gip) - Gip[i][i];   // loss_pi terms
        res[4][i] = lse_row(Gpt) - Gpt[i][i];   // loss_p  terms
    }
    __syncthreads();

    if (tid == 0) {
        float acc[5] = {0.f, 0.f, 0.f, 0.f, 0.f};
        for (int i = 0; i < 16; ++i)
            for (int k = 0; k < 5; ++k) acc[k] += res[k][i];
        const float loss_it = acc[0] / 16.f;
        const float loss_ti = acc[1] / 16.f;
        const float loss_ip = acc[2] / 16.f;
        const float loss_pi = acc[3] / 16.f;
        const float loss_p  = acc[4] / 16.f;
        const float loss_global = 0.25f * (loss_it + loss_ti + loss_ip + loss_pi);
        const float loss_cross  = 0.5f  * (loss_it + loss_p);

        // local: pos term from per-b diagonal means, neg term from partials
        float pos_loss = 0.f;
        for (int b = 0; b < BB; ++b) {
            float sdi = 0.f, sdp = 0.f;
            for (int t = 0; t < TILES; ++t) {
                sdi += ws[OFF_PDI + b * TILES + t];
                sdp += ws[OFF_PDP + b * TILES + t];
            }
            pos_loss += logf(sigmoidf_(sdi / (float)HWN) + 1e-8f);
            pos_loss += logf(sigmoidf_(sdp / (float)HWN) + 1e-8f);
        }
        pos_loss = -pos_loss / (float)(2 * BB);

        float negsum = 0.f;
        for (int k = 0; k < NBLK; ++k) negsum += ws[OFF_PNEG + k];
        const float neg_loss   = negsum / (float)(2 * BB * HWN);
        const float loss_local = pos_loss + 0.1f * neg_loss;

        const float g  = loss_global * 1.0f;
        const float ll = loss_local  * 0.5f;
        const float cx = loss_cross  * 0.5f;
        out[0] = g;
        out[1] = ll;
        out[2] = cx;
        out[3] = g + ll + cx;
    }
}

// ---------------------------------------------------------------------------
extern "C" void kernel_launch(void* const* d_in, const int* in_sizes, int n_in,
                              void* d_out, int out_size, void* d_ws, size_t ws_size,
                              hipStream_t stream)
{
    const float* ig  = (const float*)d_in[0];
    const float* pg  = (const float*)d_in[1];
    const float* tg  = (const float*)d_in[2];
    const float* imf = (const float*)d_in[3];
    const float* ptf = (const float*)d_in[4];
    const float* bvf = (const float*)d_in[5];
    const float* lsc = (const float*)d_in[6];
    float* out = (float*)d_out;
    float* ws  = (float*)d_ws;
    (void)in_sizes; (void)n_in; (void)out_size; (void)ws_size;

    k1_norms_diags<<<NBLK, 128, 0, stream>>>(imf, ptf, bvf, ws);
    k2_bevsum<<<BB * CC, 256, 0, stream>>>(bvf, ws);
    k3_neg<<<NBLK, 128, 0, stream>>>(imf, ptf, ws);
    k4_final<<<1, 256, 0, stream>>>(ig, pg, tg, lsc, ws, out);
}